// SkBertConcatClassifier_24369644438202
// MI455X (gfx1250) — compile-verified
//
#include <hip/hip_runtime.h>
#include <hip/hip_bf16.h>

typedef __attribute__((ext_vector_type(2))) float    v2f;
typedef __attribute__((ext_vector_type(8))) float    v8f;
typedef __attribute__((ext_vector_type(4))) unsigned u32x4;
typedef __attribute__((ext_vector_type(8))) unsigned u32x8;

#define Bn    32768
#define Hc    768
#define Dc    1536
#define CHUNK 128
#define NCHUNK (Dc / CHUNK)            // 12
#define ASTRIDE 132                    // 128 + 4 pad floats (TDM pad feature)
#define WMP_ELEMS (Dc * 16)            // 24576 floats: Wmoe, (K,K+1)-pair order
#define WHP_ELEMS (Dc * 4)             // 6144 floats: compact head (3 cols + ones)
#define ZERO_OFF  (WMP_ELEMS + WHP_ELEMS)  // 30720, 32 zeroed floats
#define ABASE     (ZERO_OFF + 32)      // 30752
#define ABUF      (16 * ASTRIDE)       // 2112 floats per buffer
#define LN_EPS 1e-5f

// ---------------------------------------------------------------------------
// Setup kernel 1: build pair-shuffled weight images in ws.
//   WmP index i -> j=i&1, n=(i>>1)&15, h=(i>>5)&1, kg=i>>6, d=4*kg+2*h+j
//     so a lane's B pair (W[k+kofs][n], W[k+kofs+1][n]) is one b64.
//   WhP same with 4 columns: 0..2 = gamma*head_w, 3 = 1.0 (row-sum column).
// ---------------------------------------------------------------------------
__global__ void build_weights(const float* __restrict__ router_w,
                              const float* __restrict__ expert_w,
                              const float* __restrict__ ln_gamma,
                              const float* __restrict__ head_w,
                              float* __restrict__ WmP,
                              float* __restrict__ WhP) {
    int i = blockIdx.x * blockDim.x + threadIdx.x;
    if (i < WMP_ELEMS) {
        int j = i & 1, n = (i >> 1) & 15, h = (i >> 5) & 1, kg = i >> 6;
        int d = kg * 4 + 2 * h + j;
        float v;
        if (n < 4) {
            v = router_w[d * 4 + n];
        } else {
            int e = (n - 4) / 3, l = (n - 4) % 3;
            v = expert_w[(size_t)e * (Dc * 3) + d * 3 + l];
        }
        WmP[i] = v;
    }
    if (i < WHP_ELEMS) {
        int j = i & 1, n = (i >> 1) & 3, h = (i >> 3) & 1, kg = i >> 4;
        int d = kg * 4 + 2 * h + j;
        WhP[i] = (n < 3) ? ln_gamma[d] * head_w[d * 3 + n] : 1.0f;
    }
}

// ---------------------------------------------------------------------------
// Setup kernel 2: G[l] = sum_d gamma*hw,  C[l] = sum_d beta*hw + head_b
// ---------------------------------------------------------------------------
__global__ void build_gc(const float* __restrict__ ln_gamma,
                         const float* __restrict__ ln_beta,
                         const float* __restrict__ head_w,
                         const float* __restrict__ head_b,
                         float* __restrict__ GC) {
    __shared__ float s[6][256];
    float g[3] = {0.f, 0.f, 0.f}, c[3] = {0.f, 0.f, 0.f};
    for (int d = threadIdx.x; d < Dc; d += 256) {
        float ga = ln_gamma[d], be = ln_beta[d];
#pragma unroll
        for (int l = 0; l < 3; ++l) {
            float w = head_w[d * 3 + l];
            g[l] += ga * w;
            c[l] += be * w;
        }
    }
#pragma unroll
    for (int l = 0; l < 3; ++l) {
        s[l][threadIdx.x]     = g[l];
        s[3 + l][threadIdx.x] = c[l];
    }
    __syncthreads();
    if (threadIdx.x < 6) {
        float acc = 0.f;
        for (int t = 0; t < 256; ++t) acc += s[threadIdx.x][t];
        if (threadIdx.x >= 3) acc += head_b[threadIdx.x - 3];
        GC[threadIdx.x] = acc;
    }
}

// ---------------------------------------------------------------------------
// TDM: one tensor_load_to_lds per (wave, chunk).
// D# group0: count=1 | lds_addr | global_addr | type=2
// D# group1: data_size=4B, pad_enable, pad_interval=128dw(code6),
//            pad_amount=4dw(code3); tensor/tile = 128x16 dwords,
//            tensor_dim0_stride = 768 (row stride of [B,768] activations).
// ---------------------------------------------------------------------------
__device__ __forceinline__ void tdm_issue(const float* src, int rowBase, int col,
                                          unsigned lds_off_bytes) {
    unsigned long long ga =
        (unsigned long long)(uintptr_t)(const void*)(src + (size_t)rowBase * Hc + col);
    unsigned ga_lo  = __builtin_amdgcn_readfirstlane((unsigned)ga);
    unsigned ga_hi  = __builtin_amdgcn_readfirstlane((unsigned)(ga >> 32));
    unsigned ldsoff = __builtin_amdgcn_readfirstlane(lds_off_bytes);

    u32x4 g0;
    g0[0] = 1u;                                    // count=1, user descriptor
    g0[1] = ldsoff;                                // lds_addr (bytes)
    g0[2] = ga_lo;                                 // global_addr[31:0]
    g0[3] = (ga_hi & 0x01FFFFFFu) | (2u << 30);    // global_addr[56:32] | type=2

    u32x8 g1;
    g1[0] = (2u << 16) | (1u << 20) | (6u << 22) | (3u << 25); // dsize/pad cfg
    g1[1] = (128u << 16);                          // tensor_dim0[15:0] = 128
    g1[2] = (16u << 16);                           // tensor_dim1[15:0] = 16
    g1[3] = (128u << 16);                          // tile_dim0 = 128
    g1[4] = 16u;                                   // tile_dim1 = 16, tile_dim2 = 0
    g1[5] = 768u;                                  // tensor_dim0_stride = 768
    g1[6] = 0u;
    g1[7] = 0u;

    asm volatile("tensor_load_to_lds %0, %1" :: "s"(g0), "s"(g1) : "memory");
}

// ---------------------------------------------------------------------------
// Main fused kernel: 256 threads = 8 waves; each wave owns 16 rows.
// ---------------------------------------------------------------------------
__global__ __launch_bounds__(256)
void asc_fused_kernel(const float* __restrict__ sent,
                      const float* __restrict__ term,
                      const float* __restrict__ WmP,
                      const float* __restrict__ WhP,
                      const float* __restrict__ GC,
                      const float* __restrict__ router_b,
                      const float* __restrict__ expert_b,
                      float* __restrict__ out) {
    extern __shared__ float smem[];
    float* sWm = smem;                 // pair-shuffled [Dc/4][2][16][2]
    float* sWh = smem + WMP_ELEMS;     // pair-shuffled [Dc/4][2][4][2]

    const int tid  = threadIdx.x;
    const int wave = tid >> 5;
    const int lane = tid & 31;

    // ---- cooperative preload of weights into LDS + zero slot ----
    {
        const float4* srcM = (const float4*)WmP;
        float4* dstM = (float4*)smem;
        for (int i = tid; i < (WMP_ELEMS + WHP_ELEMS) / 4; i += 256)
            dstM[i] = srcM[i];
        if (tid < 32) smem[ZERO_OFF + tid] = 0.0f;
    }
    __syncthreads();

    const int rowBase = blockIdx.x * 128 + wave * 16;
    const int n = lane & 15;
    const int h = lane >> 4;                 // half-wave: K offset 0 or 2
    const unsigned abase = ABASE + wave * (2 * ABUF);

    // ---- TDM prologue: DMA chunk 0 into buffer 0 ----
    tdm_issue(sent, rowBase, 0, abase * 4u);

    v8f cm = {};                             // router+experts (16 cols)
    v8f ch = {};                             // head cols 0-2, col 3 = row sum
    float sq = 0.0f;                         // per-lane partial sum(x^2)

    for (int c = 0; c < NCHUNK; ++c) {
        // DMA next chunk into the other buffer, overlap with this chunk's WMMA
        if (c + 1 < NCHUNK) {
            const int cb = (c + 1) * CHUNK;
            tdm_issue(cb < Hc ? sent : term, rowBase, cb < Hc ? cb : cb - Hc,
                      (abase + ((c + 1) & 1) * ABUF) * 4u);
            __builtin_amdgcn_s_wait_tensorcnt(1);   // current buffer complete
        } else {
            __builtin_amdgcn_s_wait_tensorcnt(0);
        }
        asm volatile("" ::: "memory");

        const float* bA = smem + abase + (c & 1) * ABUF;
#pragma unroll 8
        for (int kk = 0; kk < CHUNK; kk += 4) {
            const int kg = c * 32 + (kk >> 2);         // global K group
            v2f a = *(const v2f*)(bA + n * ASTRIDE + kk + 2 * h);
            sq += a.x * a.x + a.y * a.y;
            v2f bm = *(const v2f*)(sWm + ((kg * 2 + h) * 16 + n) * 2);
            const float* ph = (n < 4) ? (sWh + ((kg * 2 + h) * 4 + n) * 2)
                                      : (smem + ZERO_OFF);
            v2f bh = *(const v2f*)ph;
            cm = __builtin_amdgcn_wmma_f32_16x16x4_f32(
                    false, a, false, bm, (short)0, cm, false, false);
            ch = __builtin_amdgcn_wmma_f32_16x16x4_f32(
                    false, a, false, bh, (short)0, ch, false, false);
        }
    }

    // ---- sum(x^2): lane r and lane r+16 hold row r's two halves ----
    const float rowSq = sq + __shfl_xor(sq, 16, 32);

    // ---- dump C tiles (VGPR j -> M=j lanes 0-15, M=j+8 lanes 16-31) ----
    float* EP = smem + abase;                 // reuse buffer 0
    {
        const int mbase = h * 8;
#pragma unroll
        for (int j = 0; j < 8; ++j) {
            EP[(mbase + j) * 16 + n]       = cm[j];
            EP[256 + (mbase + j) * 16 + n] = ch[j];
        }
    }
    asm volatile("s_wait_dscnt 0" ::: "memory");

    // ---- per-row epilogue: LN fold + top-2 router softmax + combine ----
    if (lane < 16) {
        const int r = lane;
        const float rowSum = EP[256 + r * 16 + 3];   // from WMMA ones column
        const float mu   = rowSum * (1.0f / Dc);
        const float var  = rowSq * (1.0f / Dc) - mu * mu;
        const float rinv = rsqrtf(var + LN_EPS);

        float rv[4];
#pragma unroll
        for (int e = 0; e < 4; ++e) rv[e] = EP[r * 16 + e] + router_b[e];

        int i1 = 0;
#pragma unroll
        for (int e = 1; e < 4; ++e) if (rv[e] > rv[i1]) i1 = e;
        int i2 = (i1 == 0) ? 1 : 0;
#pragma unroll
        for (int e = 0; e < 4; ++e) if (e != i1 && rv[e] > rv[i2]) i2 = e;

        const float g1 = 1.0f / (1.0f + __expf(rv[i2] - rv[i1])); // rv[i1]>=rv[i2]
        const float g2 = 1.0f - g1;

#pragma unroll
        for (int l = 0; l < 3; ++l) {
            float e1   = EP[r * 16 + 4 + i1 * 3 + l] + expert_b[i1 * 3 + l];
            float e2   = EP[r * 16 + 4 + i2 * 3 + l] + expert_b[i2 * 3 + l];
            float hd   = EP[256 + r * 16 + l];
            float base = rinv * (hd - mu * GC[l]) + GC[3 + l];
            out[(size_t)(rowBase + r) * 3 + l] = base + 1.0f * (g1 * e1 + g2 * e2);
        }
    }
}

// ---------------------------------------------------------------------------
extern "C" void kernel_launch(void* const* d_in, const int* in_sizes, int n_in,
                              void* d_out, int out_size, void* d_ws, size_t ws_size,
                              hipStream_t stream) {
    const float* cls_sent = (const float*)d_in[0];
    const float* cls_term = (const float*)d_in[1];
    const float* ln_gamma = (const float*)d_in[2];
    const float* ln_beta  = (const float*)d_in[3];
    const float* head_w   = (const float*)d_in[4];
    const float* head_b   = (const float*)d_in[5];
    const float* router_w = (const float*)d_in[6];
    const float* router_b = (const float*)d_in[7];
    const float* expert_w = (const float*)d_in[8];
    const float* expert_b = (const float*)d_in[9];
    float* out = (float*)d_out;

    float* ws   = (float*)d_ws;
    float* WmP  = ws;                         // 24576 floats
    float* WhP  = ws + WMP_ELEMS;             // 6144 floats
    float* GC   = ws + WMP_ELEMS + WHP_ELEMS; // 6 floats

    build_weights<<<WMP_ELEMS / 256, 256, 0, stream>>>(
        router_w, expert_w, ln_gamma, head_w, WmP, WhP);
    build_gc<<<1, 256, 0, stream>>>(ln_gamma, ln_beta, head_w, head_b, GC);

    const size_t lds_bytes = (size_t)(ABASE + 8 * 2 * ABUF) * sizeof(float); // ~258 KB
    asc_fused_kernel<<<Bn / 128, 256, lds_bytes, stream>>>(
        cls_sent, cls_term, WmP, WhP, GC, router_b, expert_b, out);
}